// ProtGNNNet_3556232921300
// MI455X (gfx1250) — compile-verified
//
#include <hip/hip_runtime.h>
#include <hip/hip_bf16.h>
#include <math.h>

typedef __attribute__((ext_vector_type(2))) float v2f;
typedef __attribute__((ext_vector_type(8))) float v8f;

#define N_NODES  50000
#define N_EDGES  800000
#define N_GRAPHS 64
#define IN_DIM   128
#define HID      256
#define N_LAYERS 4
#define EPSV     1e-12f

__device__ __forceinline__ v8f wmma4(v2f a, v2f b, v8f c) {
  // V_WMMA_F32_16X16X4_F32 : D = A(16x4,f32) * B(4x16,f32) + C(16x16,f32)
  return __builtin_amdgcn_wmma_f32_16x16x4_f32(false, a, false, b, (short)0, c, false, false);
}

__global__ void zero_f32(float* __restrict__ p, long long n) {
  long long i = (long long)blockIdx.x * blockDim.x + threadIdx.x;
  long long stride = (long long)gridDim.x * blockDim.x;
  for (; i < n; i += stride) p[i] = 0.0f;
}

// Repack W[K][256] row-major -> float2 pairs along K: Wp2[k2*256+n] = {W[2k2][n], W[2k2+1][n]}
__global__ void repack_w(const float* __restrict__ W, float* __restrict__ Wp, int K) {
  int idx = blockIdx.x * blockDim.x + threadIdx.x;
  int total = (K >> 1) * HID;
  if (idx >= total) return;
  int n  = idx & (HID - 1);
  int k2 = idx >> 8;
  float2 v;
  v.x = W[(size_t)(2 * k2) * HID + n];
  v.y = W[(size_t)(2 * k2 + 1) * HID + n];
  ((float2*)Wp)[(size_t)k2 * HID + n] = v;
}

__global__ void count_dst(const int* __restrict__ dst, float* __restrict__ deg, int nE) {
  int i = blockIdx.x * blockDim.x + threadIdx.x;
  if (i < nE) atomicAdd(&deg[dst[i]], 1.0f);
}

__global__ void finish_deg(float* __restrict__ deg, int n) {
  int i = blockIdx.x * blockDim.x + threadIdx.x;
  if (i < n) deg[i] = 1.0f / fmaxf(deg[i], 1.0f);   // invdeg, empty segments -> divide by 1
}

// One wave per edge: gather x[src] (2x float4 per lane) and atomically add into cacc[dst].
__global__ void edge_agg(const float* __restrict__ x, const int* __restrict__ src,
                         const int* __restrict__ dst, float* __restrict__ cacc, int nE) {
  int wave = (int)(((long long)blockIdx.x * blockDim.x + threadIdx.x) >> 5);
  int lane = threadIdx.x & 31;
  if (wave >= nE) return;
  int s = src[wave];
  int d = dst[wave];
  const float4* xs = (const float4*)(x + (size_t)s * HID);
  float4 v0 = xs[lane];
  float4 v1 = xs[lane + 32];
  float* cd = cacc + (size_t)d * HID;
  atomicAdd(cd + lane * 4 + 0, v0.x);
  atomicAdd(cd + lane * 4 + 1, v0.y);
  atomicAdd(cd + lane * 4 + 2, v0.z);
  atomicAdd(cd + lane * 4 + 3, v0.w);
  atomicAdd(cd + 128 + lane * 4 + 0, v1.x);
  atomicAdd(cd + 128 + lane * 4 + 1, v1.y);
  atomicAdd(cd + 128 + lane * 4 + 2, v1.z);
  atomicAdd(cd + 128 + lane * 4 + 3, v1.w);
}

// WMMA GEMM. 256 threads = 8 waves; each wave computes a 16x64 strip (4 C tiles sharing
// one A fragment); block covers 32 rows x 256 cols.
// !LAYER: X[row] = h @ Wp + bias                           (A0 = h,  K = KD)
//  LAYER: bundle = [x | cacc*invdeg] @ Wp + bias; then
//         X[row] += relu(bundle / max(||bundle||, eps))    (A0 = x, A1 = cacc, K = 512)
template <int KD, bool LAYER>
__global__ void __launch_bounds__(256)
wmma_gemm(const float* __restrict__ A0, const float* __restrict__ A1,
          const float* __restrict__ invdeg, const float* __restrict__ Wp,
          const float* __restrict__ bias, float* __restrict__ X, int M) {
  __shared__ float lds[LAYER ? 32 * HID : 1];
  __shared__ float norms[LAYER ? 32 : 1];

  const int lane    = threadIdx.x & 31;
  const int wv      = threadIdx.x >> 5;
  const int rowTile = wv >> 2;                 // 0..1
  const int colBase = (wv & 3) << 6;           // 0,64,128,192
  const int r0      = blockIdx.x * 32 + (rowTile << 4);
  int rowA = r0 + (lane & 15);
  if (rowA > M - 1) rowA = M - 1;              // clamp stays inside this block's window
  const int khalf = (lane >> 4) << 1;          // lanes 0-15: K+0,1 ; lanes 16-31: K+2,3
  const int n0    = colBase + (lane & 15);
  const float2* Wp2 = (const float2*)Wp;

  v8f acc[4] = {};

  {  // phase 1: A from A0 (h for emb, x for layers)
    const int strideA = LAYER ? HID : KD;
    const float* arow = A0 + (size_t)rowA * strideA + khalf;
    const int K1 = LAYER ? HID : KD;
    for (int k = 0; k < K1; k += 4) {
      v2f a = *(const v2f*)(arow + k);
      int k2 = (k >> 1) + (lane >> 4);
      const float2* wb = Wp2 + (size_t)k2 * HID + n0;
      float2 w0 = wb[0], w1 = wb[16], w2 = wb[32], w3 = wb[48];
      v2f b0, b1, b2, b3;
      b0.x = w0.x; b0.y = w0.y;  b1.x = w1.x; b1.y = w1.y;
      b2.x = w2.x; b2.y = w2.y;  b3.x = w3.x; b3.y = w3.y;
      acc[0] = wmma4(a, b0, acc[0]);
      acc[1] = wmma4(a, b1, acc[1]);
      acc[2] = wmma4(a, b2, acc[2]);
      acc[3] = wmma4(a, b3, acc[3]);
    }
  }
  if (LAYER) {  // phase 2: A = cacc * invdeg (mean aggregation half of the concat)
    const float s = invdeg[rowA];
    const float* crow = A1 + (size_t)rowA * HID + khalf;
    for (int k = 0; k < HID; k += 4) {
      v2f a = *(const v2f*)(crow + k);
      a.x *= s; a.y *= s;
      int k2 = ((k + HID) >> 1) + (lane >> 4);
      const float2* wb = Wp2 + (size_t)k2 * HID + n0;
      float2 w0 = wb[0], w1 = wb[16], w2 = wb[32], w3 = wb[48];
      v2f b0, b1, b2, b3;
      b0.x = w0.x; b0.y = w0.y;  b1.x = w1.x; b1.y = w1.y;
      b2.x = w2.x; b2.y = w2.y;  b3.x = w3.x; b3.y = w3.y;
      acc[0] = wmma4(a, b0, acc[0]);
      acc[1] = wmma4(a, b1, acc[1]);
      acc[2] = wmma4(a, b2, acc[2]);
      acc[3] = wmma4(a, b3, acc[3]);
    }
  }

  // C tile layout: VGPR r -> lanes 0-15:(M=r, N=lane) ; lanes 16-31:(M=r+8, N=lane-16)
  if (!LAYER) {
    const int rbase = r0 + ((lane >> 4) << 3);
#pragma unroll
    for (int t = 0; t < 4; ++t) {
      int col = colBase + (t << 4) + (lane & 15);
      float bv = bias[col];
#pragma unroll
      for (int r = 0; r < 8; ++r) {
        int row = rbase + r;
        if (row < M) X[(size_t)row * HID + col] = acc[t][r] + bv;
      }
    }
  } else {
    const int rb = (rowTile << 4) + ((lane >> 4) << 3);
#pragma unroll
    for (int t = 0; t < 4; ++t) {
      int col = colBase + (t << 4) + (lane & 15);
      float bv = bias[col];
#pragma unroll
      for (int r = 0; r < 8; ++r) lds[(rb + r) * HID + col] = acc[t][r] + bv;
    }
    __syncthreads();
    if (threadIdx.x < 32) norms[threadIdx.x] = 0.0f;
    __syncthreads();
    const int lrow = threadIdx.x >> 3;
    const int c0   = (threadIdx.x & 7) << 5;
    float ssum = 0.0f;
#pragma unroll
    for (int j = 0; j < 32; ++j) { float v = lds[lrow * HID + c0 + j]; ssum += v * v; }
    atomicAdd(&norms[lrow], ssum);
    __syncthreads();
    const int g = blockIdx.x * 32 + lrow;
    if (g < M) {
      float inv = 1.0f / fmaxf(sqrtf(norms[lrow]), EPSV);
      float* xr = X + (size_t)g * HID + c0;
#pragma unroll
      for (int j = 0; j < 32; ++j) {
        float v = lds[lrow * HID + c0 + j] * inv;
        xr[j] += fmaxf(v, 0.0f);
      }
    }
  }
}

// One wave per node: atomic-accumulate node features into its graph slot.
__global__ void graph_pool(const float* __restrict__ x, const int* __restrict__ gid,
                           float* __restrict__ hg, float* __restrict__ gcnt, int N) {
  int node = (int)(((long long)blockIdx.x * blockDim.x + threadIdx.x) >> 5);
  int lane = threadIdx.x & 31;
  if (node >= N) return;
  int g = gid[node];
  const float4* xs = (const float4*)(x + (size_t)node * HID);
  float4 v0 = xs[lane];
  float4 v1 = xs[lane + 32];
  float* hgd = hg + (size_t)g * HID;
  atomicAdd(hgd + lane * 4 + 0, v0.x);
  atomicAdd(hgd + lane * 4 + 1, v0.y);
  atomicAdd(hgd + lane * 4 + 2, v0.z);
  atomicAdd(hgd + lane * 4 + 3, v0.w);
  atomicAdd(hgd + 128 + lane * 4 + 0, v1.x);
  atomicAdd(hgd + 128 + lane * 4 + 1, v1.y);
  atomicAdd(hgd + 128 + lane * 4 + 2, v1.z);
  atomicAdd(hgd + 128 + lane * 4 + 3, v1.w);
  if (lane == 0) atomicAdd(&gcnt[g], 1.0f);
}

// One block per graph: prototype L2 distances, log-similarity, FC, sigmoid.
__global__ void head(const float* __restrict__ hg, const float* __restrict__ gcnt,
                     const float* __restrict__ ppos, const float* __restrict__ pneg,
                     const float* __restrict__ wfc, float* __restrict__ out) {
  __shared__ float d2[10];
  const int g = blockIdx.x;
  const int t = threadIdx.x;           // 256 threads == HID
  if (t < 10) d2[t] = 0.0f;
  __syncthreads();
  float inv = 1.0f / fmaxf(gcnt[g], 1.0f);
  float hv = hg[(size_t)g * HID + t] * inv;
#pragma unroll
  for (int p = 0; p < 5; ++p) {
    float a = hv - ppos[p * HID + t];
    atomicAdd(&d2[p], a * a);
    float b = hv - pneg[p * HID + t];
    atomicAdd(&d2[5 + p], b * b);
  }
  __syncthreads();
  if (t == 0) {
    float y = 0.0f;
#pragma unroll
    for (int p = 0; p < 10; ++p) {
      float d = d2[p];
      y += logf((d + 1.0f) / (d + EPSV)) * wfc[p];
    }
    out[g] = 1.0f / (1.0f + expf(-y));
  }
}

extern "C" void kernel_launch(void* const* d_in, const int* in_sizes, int n_in,
                              void* d_out, int out_size, void* d_ws, size_t ws_size,
                              hipStream_t stream) {
  const float* h    = (const float*)d_in[0];
  const int*   src  = (const int*)d_in[1];
  const int*   dst  = (const int*)d_in[2];
  const int*   gid  = (const int*)d_in[3];
  const float* Wemb = (const float*)d_in[4];
  const float* bemb = (const float*)d_in[5];
  const float* Ws   = (const float*)d_in[6];
  const float* bs   = (const float*)d_in[7];
  const float* ppos = (const float*)d_in[8];
  const float* pneg = (const float*)d_in[9];
  const float* wfc  = (const float*)d_in[10];
  float* out = (float*)d_out;

  float* ws = (float*)d_ws;
  size_t off = 0;
  float* x    = ws + off; off += (size_t)N_NODES * HID;          // 12.8M
  float* cacc = ws + off; off += (size_t)N_NODES * HID;          // 12.8M
  float* deg  = ws + off; off += 50048;                          // invdeg (padded)
  float* hg   = ws + off; off += (size_t)N_GRAPHS * HID;
  float* gcnt = ws + off; off += 64;
  float* WpE  = ws + off; off += (size_t)IN_DIM * HID;           // packed W_emb
  float* WpL  = ws + off; off += (size_t)N_LAYERS * 2 * HID * HID; // packed Ws

  // zero deg + hg + gcnt (contiguous region)
  zero_f32<<<256, 256, 0, stream>>>(deg, 50048LL + (long long)N_GRAPHS * HID + 64);
  count_dst<<<(N_EDGES + 255) / 256, 256, 0, stream>>>(dst, deg, N_EDGES);
  finish_deg<<<(N_NODES + 255) / 256, 256, 0, stream>>>(deg, N_NODES);

  repack_w<<<((IN_DIM / 2) * HID + 255) / 256, 256, 0, stream>>>(Wemb, WpE, IN_DIM);
  for (int l = 0; l < N_LAYERS; ++l)
    repack_w<<<((2 * HID / 2) * HID + 255) / 256, 256, 0, stream>>>(
        Ws + (size_t)l * 2 * HID * HID, WpL + (size_t)l * 2 * HID * HID, 2 * HID);

  const int gemmBlocks = (N_NODES + 31) / 32;
  wmma_gemm<IN_DIM, false><<<gemmBlocks, 256, 0, stream>>>(h, nullptr, nullptr, WpE, bemb, x, N_NODES);

  for (int l = 0; l < N_LAYERS; ++l) {
    zero_f32<<<4096, 256, 0, stream>>>(cacc, (long long)N_NODES * HID);
    edge_agg<<<N_EDGES / 8, 256, 0, stream>>>(x, src, dst, cacc, N_EDGES);
    wmma_gemm<2 * HID, true><<<gemmBlocks, 256, 0, stream>>>(
        x, cacc, deg, WpL + (size_t)l * 2 * HID * HID, bs + (size_t)l * HID, x, N_NODES);
  }

  graph_pool<<<(N_NODES * 32 + 255) / 256, 256, 0, stream>>>(x, gid, hg, gcnt, N_NODES);
  head<<<N_GRAPHS, 256, 0, stream>>>(hg, gcnt, ppos, pneg, wfc, out);
}